// PositionEncoding_14508399526634
// MI455X (gfx1250) — compile-verified
//
#include <hip/hip_runtime.h>
#include <hip/hip_bf16.h>
#include <stdint.h>

// Problem constants (match reference): B=4, N=4096, D=64, K=16
#define BN 4
#define NN 4096
#define DD 64
#define KTOP 16
#define CHUNK 512       // distance columns buffered per LDS chunk in knn kernel
#define MLP_WAVES 4     // waves (== points) per block in mlp kernel

typedef float v2f __attribute__((ext_vector_type(2)));
typedef float v8f __attribute__((ext_vector_type(8)));

// D(16x16,f32) = A(16x4,f32) * B(4x16,f32) + C  -- CDNA5 V_WMMA_F32_16X16X4_F32
__device__ __forceinline__ v8f wmma_f32_16x16x4(v2f a, v2f b, v8f c) {
#if __has_builtin(__builtin_amdgcn_wmma_f32_16x16x4_f32)
  return __builtin_amdgcn_wmma_f32_16x16x4_f32(
      /*neg_a=*/false, a, /*neg_b=*/false, b,
      /*c_mod=*/(short)0, c, /*reuse_a=*/false, /*reuse_b=*/false);
#else
  (void)a; (void)b;
  return c;
#endif
}

// ---------------------------------------------------------------- sq norms
__global__ __launch_bounds__(256) void sqnorm_kernel(const float* __restrict__ x,
                                                     float* __restrict__ sq) {
  int i = blockIdx.x * blockDim.x + threadIdx.x;   // 0 .. B*N-1
  const float4* xr = (const float4*)(x + (size_t)i * DD);
  float s = 0.f;
#pragma unroll
  for (int j = 0; j < DD / 4; ++j) {
    float4 v = xr[j];
    s += v.x * v.x + v.y * v.y + v.z * v.z + v.w * v.w;
  }
  sq[i] = s;
}

// --------------------------------------------------- fused distances + topk
// One wave (32 lanes) per (batch, 16-row tile). WMMA computes x.x^T tiles
// (two independent K-split accumulator chains to avoid WMMA RAW stalls),
// d2 chunks staged in LDS. Each row's candidate scan is split across the lane
// pair (r, r+16) by column parity (bank-conflict-free), each keeping a private
// sorted top-17 list; the two lists merge lexicographically at the end.
__global__ __launch_bounds__(32) void knn_kernel(const float* __restrict__ x,
                                                 const float* __restrict__ sq,
                                                 int* __restrict__ idxo) {
  __shared__ float dchunk[16][CHUNK + 2];   // padded: conflict-free scan
  __shared__ float sqAs[16];
  __shared__ float mD[16][KTOP + 1];
  __shared__ int   mI[16][KTOP + 1];

  const int lane = threadIdx.x;
  const int tiles_per_b = NN / 16;
  const int b = blockIdx.x / tiles_per_b;
  const int row_base = (blockIdx.x % tiles_per_b) * 16;

  const float* xb  = x  + (size_t)b * NN * DD;
  const float* sqb = sq + (size_t)b * NN;

  const int mr   = lane & 15;                 // M (A) / N (B) index within tile
  const int half = lane >> 4;                 // scan parity for this lane
  const int ksel = half * 2;                  // K sub-select per lane half
  const int moff = half * 8;                  // C-layout row offset

  if (lane < 16) sqAs[lane] = sqb[row_base + lane];
  __syncthreads();

  // Preload A fragments for the whole row tile: 16 k-steps of 16x4 f32.
  v2f afrag[16];
#pragma unroll
  for (int kk = 0; kk < 16; ++kk) {
    const float* ap = xb + (size_t)(row_base + mr) * DD + kk * 4 + ksel;
    afrag[kk].x = ap[0];
    afrag[kk].y = ap[1];
  }

  float bestd[KTOP + 1];
  int   besti[KTOP + 1];
#pragma unroll
  for (int t = 0; t <= KTOP; ++t) { bestd[t] = 3.4e38f; besti[t] = 0; }
  float thr = 3.4e38f;                        // register copy of bestd[KTOP]

  for (int cb = 0; cb < NN; cb += CHUNK) {
    for (int ct = 0; ct < CHUNK / 16; ++ct) {
      const int col_base = cb + ct * 16;
      if (ct + 2 < CHUNK / 16)  // hide L2->L0 latency for the tile after next
        __builtin_prefetch(xb + (size_t)(col_base + 32 + mr) * DD, 0, 0);
      // Two independent accumulator chains (K split) -> interleavable WMMAs.
      v8f acc0 = {0.f, 0.f, 0.f, 0.f, 0.f, 0.f, 0.f, 0.f};
      v8f acc1 = {0.f, 0.f, 0.f, 0.f, 0.f, 0.f, 0.f, 0.f};
      const float* bp = xb + (size_t)(col_base + mr) * DD + ksel;
#pragma unroll
      for (int kk = 0; kk < 8; ++kk) {
        v2f bf0; bf0.x = bp[kk * 4];       bf0.y = bp[kk * 4 + 1];
        v2f bf1; bf1.x = bp[(kk + 8) * 4]; bf1.y = bp[(kk + 8) * 4 + 1];
        acc0 = wmma_f32_16x16x4(afrag[kk], bf0, acc0);
        acc1 = wmma_f32_16x16x4(afrag[kk + 8], bf1, acc1);
      }
      const float sqc = sqb[col_base + mr];   // ||x_col||^2, N = lane%16
#pragma unroll
      for (int v = 0; v < 8; ++v) {
        const int M = v + moff;
        dchunk[M][ct * 16 + mr] = sqAs[M] + sqc - 2.0f * (acc0[v] + acc1[v]);
      }
    }
    __syncthreads();

    // Interleaved scan: lane (r) takes even columns, lane (r+16) odd columns.
    for (int j = 0; j < CHUNK / 2; ++j) {
      const int jj = (j << 1) | half;
      const float d = dchunk[mr][jj];
      if (d < thr) {
        const int col = cb + jj;
        int p = KTOP;
        while (p > 0 && bestd[p - 1] > d) {
          bestd[p] = bestd[p - 1];
          besti[p] = besti[p - 1];
          --p;
        }
        bestd[p] = d;
        besti[p] = col;
        thr = bestd[KTOP];
      }
    }
    __syncthreads();
  }

  // Merge the odd-lane lists into the even-lane lists (lexicographic ties).
  if (lane >= 16) {
#pragma unroll
    for (int t = 0; t <= KTOP; ++t) { mD[mr][t] = bestd[t]; mI[mr][t] = besti[t]; }
  }
  __syncthreads();
  if (lane < 16) {
    for (int t = 0; t <= KTOP; ++t) {
      const float d = mD[mr][t];
      const int  ci = mI[mr][t];
      if (d < bestd[KTOP] || (d == bestd[KTOP] && ci < besti[KTOP])) {
        int p = KTOP;
        while (p > 0 &&
               (bestd[p - 1] > d || (bestd[p - 1] == d && besti[p - 1] > ci))) {
          bestd[p] = bestd[p - 1];
          besti[p] = besti[p - 1];
          --p;
        }
        bestd[p] = d;
        besti[p] = ci;
      }
    }
    int* op = idxo + ((size_t)b * NN + row_base + lane) * KTOP;
#pragma unroll
    for (int t = 0; t < KTOP; ++t) op[t] = besti[t + 1];  // drop self (slot 0)
  }
}

// ------------------------------------------------------------ edge MLP
// One wave per point: diff(16x64) @ W1(64x64) -> relu -> @ W2(64x64) + biases.
// Weights staged TRANSPOSED in LDS so each B fragment is one ds_load_b64;
// A fragments preloaded to registers and reused across all 4 N-tiles;
// K split into two accumulator chains per N-tile.
__global__ __launch_bounds__(32 * MLP_WAVES) void mlp_kernel(
    const float* __restrict__ x, const int* __restrict__ idxi,
    const float* __restrict__ W1, const float* __restrict__ b1,
    const float* __restrict__ W2, const float* __restrict__ b2,
    float* __restrict__ out) {
  __shared__ float Wt1[64][66];                  // Wt1[c][k] = W1[k][c]
  __shared__ float Wt2[64][66];
  __shared__ float buf[MLP_WAVES][16][68];       // padded per-wave staging

  const int tid = threadIdx.x;
  for (int i = tid; i < 64 * 64; i += 32 * MLP_WAVES) {
    const int k = i >> 6, c = i & 63;            // global reads coalesced
    Wt1[c][k] = W1[i];
    Wt2[c][k] = W2[i];
  }
  __syncthreads();

  const int wave = tid >> 5;
  const int lane = tid & 31;
  const int p = blockIdx.x * MLP_WAVES + wave;    // 0 .. B*N-1
  const int row  = lane & 15;
  const int ksel = (lane < 16) ? 0 : 2;
  const int moff = (lane >= 16) ? 8 : 0;

  const int b = p >> 12;                          // p / N
  const int nb = idxi[(size_t)p * KTOP + row];    // this lane's neighbor row

  const float* xn = x + (size_t)p * DD;
  const float* xm = x + ((size_t)b * NN + nb) * DD;
  float (*mybuf)[68] = buf[wave];

  // Stage diff = x_center - x_neighbor : 16 rows x 64 cols
  {
    const int cbase = (lane < 16) ? 0 : 32;
    const float4* a4 = (const float4*)(xn + cbase);
    const float4* c4 = (const float4*)(xm + cbase);
#pragma unroll
    for (int j = 0; j < 8; ++j) {
      float4 av = a4[j], cv = c4[j];
      float4 dv = {av.x - cv.x, av.y - cv.y, av.z - cv.z, av.w - cv.w};
      *(float4*)&mybuf[row][cbase + j * 4] = dv;
    }
  }

  // Preload diff A fragments once; reuse across all 4 N-tiles.
  v2f afr[16];
#pragma unroll
  for (int kk = 0; kk < 16; ++kk)
    afr[kk] = *(const v2f*)&mybuf[row][kk * 4 + ksel];

  // h = relu(diff @ W1 + b1)  (C layout: M = v+moff, N = lane%16)
  v8f h[4];
#pragma unroll
  for (int nt = 0; nt < 4; ++nt) {
    const float bias = b1[nt * 16 + row];
    v8f acc0 = {bias, bias, bias, bias, bias, bias, bias, bias};
    v8f acc1 = {0.f, 0.f, 0.f, 0.f, 0.f, 0.f, 0.f, 0.f};
    const float* wcol = &Wt1[nt * 16 + row][0];
#pragma unroll
    for (int kk = 0; kk < 8; ++kk) {
      v2f w0 = *(const v2f*)&wcol[kk * 4 + ksel];         // one b64 LDS load
      v2f w1 = *(const v2f*)&wcol[(kk + 8) * 4 + ksel];
      acc0 = wmma_f32_16x16x4(afr[kk], w0, acc0);
      acc1 = wmma_f32_16x16x4(afr[kk + 8], w1, acc1);
    }
#pragma unroll
    for (int v = 0; v < 8; ++v) acc0[v] = fmaxf(acc0[v] + acc1[v], 0.0f);
    h[nt] = acc0;
  }

  // Re-fragment h as A matrix via LDS (C layout -> row-major -> A layout)
#pragma unroll
  for (int nt = 0; nt < 4; ++nt)
#pragma unroll
    for (int v = 0; v < 8; ++v)
      mybuf[v + moff][nt * 16 + row] = h[nt][v];

#pragma unroll
  for (int kk = 0; kk < 16; ++kk)
    afr[kk] = *(const v2f*)&mybuf[row][kk * 4 + ksel];

  // out = h @ W2 + b2, scatter per C layout
  float* op = out + (size_t)p * 16 * 64;
#pragma unroll
  for (int nt = 0; nt < 4; ++nt) {
    const float bias = b2[nt * 16 + row];
    v8f acc0 = {bias, bias, bias, bias, bias, bias, bias, bias};
    v8f acc1 = {0.f, 0.f, 0.f, 0.f, 0.f, 0.f, 0.f, 0.f};
    const float* wcol = &Wt2[nt * 16 + row][0];
#pragma unroll
    for (int kk = 0; kk < 8; ++kk) {
      v2f w0 = *(const v2f*)&wcol[kk * 4 + ksel];
      v2f w1 = *(const v2f*)&wcol[(kk + 8) * 4 + ksel];
      acc0 = wmma_f32_16x16x4(afr[kk], w0, acc0);
      acc1 = wmma_f32_16x16x4(afr[kk + 8], w1, acc1);
    }
#pragma unroll
    for (int v = 0; v < 8; ++v)
      op[(size_t)(v + moff) * 64 + nt * 16 + row] = acc0[v] + acc1[v];
  }
}

// ------------------------------------------------------------------ launch
extern "C" void kernel_launch(void* const* d_in, const int* in_sizes, int n_in,
                              void* d_out, int out_size, void* d_ws, size_t ws_size,
                              hipStream_t stream) {
  const float* x  = (const float*)d_in[0];
  const float* W1 = (const float*)d_in[1];
  const float* b1 = (const float*)d_in[2];
  const float* W2 = (const float*)d_in[3];
  const float* b2 = (const float*)d_in[4];
  // d_in[5] is k (==16), fixed at compile time.
  float* out = (float*)d_out;

  float* sqws  = (float*)d_ws;                                  // B*N floats
  int*   idxws = (int*)((char*)d_ws + (size_t)BN * NN * sizeof(float)); // B*N*K ints

  sqnorm_kernel<<<(BN * NN) / 256, 256, 0, stream>>>(x, sqws);
  knn_kernel<<<BN * (NN / 16), 32, 0, stream>>>(x, sqws, idxws);
  mlp_kernel<<<(BN * NN) / MLP_WAVES, 32 * MLP_WAVES, 0, stream>>>(
      x, idxws, W1, b1, W2, b2, out);
}